// RayleighSommerfeldPropagation_53523882443241
// MI455X (gfx1250) — compile-verified
//
#include <hip/hip_runtime.h>
#include <math.h>

typedef __attribute__((ext_vector_type(2))) float v2f;
typedef __attribute__((ext_vector_type(8))) float v8f;

#define RS_SIZE   64
#define RS_N2     4096
#define RS_NWAVES 8
#define TDIM      67              // 4 (tile span) + 63 relative offsets per axis
#define TENT      (TDIM * TDIM)   // 4489 Toeplitz entries per block

// Branch-free sincos for large-but-bounded phase (ph < 2^19 rad).
// One f64 Cody-Waite step: n < 2^17 -> abs reduction error ~1e-10 rad.
// Reduced |rem| <= pi -> raw v_sin_f32 / v_cos_f32, no slow path, no divergence.
__device__ __forceinline__ void rs_sincos(float ph, float* sn, float* cs) {
    const double d = (double)ph;
    const double n = rint(d * 0.15915494309189533576888376337251);   // 1/(2*pi)
    const float rem = (float)fma(n, -6.283185307179586476925286766559, d);
    *sn = __sinf(rem);
    *cs = __cosf(rem);
}

__global__ __launch_bounds__(256) void RayleighSommerfeld_wmma_kernel(
    const float* __restrict__ u_real,
    const float* __restrict__ u_imag,
    float* __restrict__ out)
{
    const float PX    = 4e-6f;
    const float Z     = 0.03f;
    const float Z2    = (float)(0.03 * 0.03);
    const float KW    = (float)(2.0 * M_PI / 5.32e-7);       // 2*pi/lambda
    const float C_PXL = (float)(4e-6 * 4e-6 / 5.32e-7);      // px^2/lambda

    // SoA planes so the j=0/j=1 lookups are adjacent floats within a plane
    // (single ds_load_2addr_b32 each, no operand repacking).
    __shared__ float tab_r[TENT];              // 18 KB: Re(K), mirrored index
    __shared__ float tab_i[TENT];              // 18 KB: Im(K), mirrored index
    __shared__ float red[RS_NWAVES * 512];     // 16 KB: per-wave partial tiles

    const int tid    = threadIdx.x;
    const int wave   = tid >> 5;
    const int lane   = tid & 31;
    const int half   = lane >> 4;     // selects K pair {0,1} vs {2,3}
    const int lane16 = lane & 15;     // = A-matrix M (batch) = B-matrix N (target)

    // 4x4 spatial target tile for this block
    const int tyb = (blockIdx.x >> 4) * 4;
    const int txb = (blockIdx.x & 15) * 4;

    // ---- Phase 1: fill per-block Toeplitz kernel table in LDS ----
    // Mirrored index: entry (v2,u2) <-> dx_int = txb + 3 - u2, dy_int = tyb + 3 - v2,
    // so that idx = (sy - ay + 3)*67 + (sx - bx + 3) is increasing in s.
    // K = amp*sin(kr) - i*amp*cos(kr), amp = (px^2/lambda)*(z/r2)
    for (int i = tid; i < TENT; i += 256) {
        const int   v2 = i / TDIM;
        const int   u2 = i - v2 * TDIM;
        const float dx = (float)(txb + 3 - u2) * PX;
        const float dy = (float)(tyb + 3 - v2) * PX;
        const float r2 = __fadd_rn(
            __fadd_rn(__fmul_rn(dx, dx), __fmul_rn(dy, dy)), Z2);
        const float r   = sqrtf(r2);
        const float amp = C_PXL * (Z / r2);    // precise div; amortized
        float sn, cs;
        rs_sincos(KW * r, &sn, &cs);
        tab_r[i] = amp * sn;                   //  Re(K)
        tab_i[i] = -(amp * cs);                //  Im(K)
    }
    __syncthreads();

    // ---- Phase 2: WMMA complex GEMM with LDS table lookups ----
    const int ay = lane16 >> 2;   // target y within 4x4 tile
    const int bx = lane16 & 3;    // target x within 4x4 tile
    const int CL = (3 - ay) * TDIM + (3 - bx);   // lane-constant index offset

    const float* __restrict__ ar_base = u_real + lane16 * RS_N2;  // batch = lane16
    const float* __restrict__ ai_base = u_imag + lane16 * RS_N2;

    v8f acc_r = {0.f, 0.f, 0.f, 0.f, 0.f, 0.f, 0.f, 0.f};
    v8f acc_i = {0.f, 0.f, 0.f, 0.f, 0.f, 0.f, 0.f, 0.f};

    const int sbeg = wave * (RS_N2 / RS_NWAVES);   // 512-source slice per wave

    for (int it = 0; it < (RS_N2 / RS_NWAVES) / 4; ++it) {
        const int s = sbeg + it * 4 + 2 * half;    // this lane's K pair base (even)

        // A operands: two contiguous f32 (8B aligned: s is even)
        v2f a_r = *(const v2f*)(ar_base + s);
        v2f a_i = *(const v2f*)(ai_base + s);

        // B operands: adjacent-pair lookups (s even -> no row wrap inside pair):
        // idx = (sy - ay + 3)*67 + (sx - bx + 3) = s + 3*(s>>6) + CL
        const int idx = s + 3 * (s >> 6) + CL;
        v2f b_r, b_ci;
        b_r[0]  = tab_r[idx];
        b_r[1]  = tab_r[idx + 1];
        b_ci[0] = tab_i[idx];
        b_ci[1] = tab_i[idx + 1];
        const v2f b_cp = -b_ci;   // -Im(K) = amp*cos

        // Complex MAC via 4 real WMMAs:
        // acc_r += Ar*Kr + Ai*(-Ki);  acc_i += Ar*Ki + Ai*Kr
        acc_r = __builtin_amdgcn_wmma_f32_16x16x4_f32(false, a_r, false, b_r,
                                                      (short)0, acc_r, false, false);
        acc_r = __builtin_amdgcn_wmma_f32_16x16x4_f32(false, a_i, false, b_cp,
                                                      (short)0, acc_r, false, false);
        acc_i = __builtin_amdgcn_wmma_f32_16x16x4_f32(false, a_r, false, b_ci,
                                                      (short)0, acc_i, false, false);
        acc_i = __builtin_amdgcn_wmma_f32_16x16x4_f32(false, a_i, false, b_r,
                                                      (short)0, acc_i, false, false);
    }

    // ---- Phase 3: cross-wave reduction and store ----
    // D layout: VGPR r, lane l -> M = r + 8*half, N = lane16
    float* myred = red + wave * 512;
#pragma unroll
    for (int r = 0; r < 8; ++r) {
        const int idx = (r + 8 * half) * 16 + lane16;   // M*16 + N
        myred[idx]       = acc_r[r];
        myred[256 + idx] = acc_i[r];
    }
    __syncthreads();

    // Each thread owns one (batch M, tile-target n) pair.
    const int M = tid >> 4;
    const int n = tid & 15;
    float sr = 0.f, si = 0.f;
#pragma unroll
    for (int w = 0; w < RS_NWAVES; ++w) {
        sr += red[w * 512 + tid];
        si += red[w * 512 + 256 + tid];
    }
    const int tg   = (tyb + (n >> 2)) * RS_SIZE + txb + (n & 3);  // global target
    const int gidx = M * RS_N2 + tg;                              // complex index
    out[2 * gidx]     = sr;
    out[2 * gidx + 1] = si;
}

extern "C" void kernel_launch(void* const* d_in, const int* in_sizes, int n_in,
                              void* d_out, int out_size, void* d_ws, size_t ws_size,
                              hipStream_t stream) {
    (void)in_sizes; (void)n_in; (void)d_ws; (void)ws_size; (void)out_size;
    const float* u_real = (const float*)d_in[0];
    const float* u_imag = (const float*)d_in[1];
    float* out = (float*)d_out;
    // 256 blocks = 16x16 grid of 4x4 target tiles; 256 threads = 8 waves splitting K
    RayleighSommerfeld_wmma_kernel<<<256, 256, 0, stream>>>(u_real, u_imag, out);
}